// MCMHEdgeDecoder_69681549410498
// MI455X (gfx1250) — compile-verified
//
#include <hip/hip_runtime.h>
#include <hip/hip_bf16.h>

typedef __attribute__((ext_vector_type(2))) float v2f;
typedef __attribute__((ext_vector_type(8))) float v8f;

#define D_IN 128

// ---------------- Pass 0: zero degree histogram ----------------
__global__ void sage_zero_deg(int* __restrict__ deg, int n) {
    int i = blockIdx.x * blockDim.x + threadIdx.x;
    if (i < n) deg[i] = 0;
}

// ---------------- Pass 1: histogram of destination nodes ----------------
__global__ void sage_hist(const long long* __restrict__ edges, int* __restrict__ deg,
                          int n_edges) {
    int e = blockIdx.x * blockDim.x + threadIdx.x;
    if (e < n_edges) {
        int dst = (int)edges[(long long)n_edges + e];  // row 1 = dst
        atomicAdd(&deg[dst], 1);
    }
}

// ---------------- Pass 2: single-block exclusive scan (50K elems) ----------------
__global__ void sage_scan(const int* __restrict__ deg, int* __restrict__ offs,
                          int* __restrict__ cursor, int n) {
    __shared__ int s[1024];
    __shared__ int carry_s;
    int t = threadIdx.x;
    if (t == 0) carry_s = 0;
    __syncthreads();
    for (int base = 0; base < n; base += 1024) {
        int i = base + t;
        int v = (i < n) ? deg[i] : 0;
        s[t] = v;
        __syncthreads();
        // Hillis-Steele inclusive scan over 1024
        for (int off = 1; off < 1024; off <<= 1) {
            int x = (t >= off) ? s[t - off] : 0;
            __syncthreads();
            s[t] += x;
            __syncthreads();
        }
        int incl = s[t];
        int carry = carry_s;
        if (i < n) {
            int excl = carry + incl - v;
            offs[i] = excl;
            cursor[i] = excl;
        }
        __syncthreads();
        if (t == 0) carry_s = carry + s[1023];
        __syncthreads();
    }
}

// ---------------- Pass 3: bucket fill (CSR by destination) ----------------
__global__ void sage_fill(const long long* __restrict__ edges, int* __restrict__ cursor,
                          int* __restrict__ csr_src, int n_edges) {
    int e = blockIdx.x * blockDim.x + threadIdx.x;
    if (e < n_edges) {
        int src = (int)edges[e];
        int dst = (int)edges[(long long)n_edges + e];
        int pos = atomicAdd(&cursor[dst], 1);
        csr_src[pos] = src;
    }
}

// ---------------- Pass 4: streaming gather max-aggregation ----------------
// One wave32 per node; lane L holds features 4L..4L+3 (coalesced float4 rows).
__global__ void sage_maxagg(const float* __restrict__ X, const int* __restrict__ offs,
                            const int* __restrict__ deg, const int* __restrict__ csr_src,
                            float* __restrict__ agg, int n_nodes) {
    int wave = (blockIdx.x * blockDim.x + threadIdx.x) >> 5;
    int lane = threadIdx.x & 31;
    if (wave >= n_nodes) return;
    int beg = offs[wave];
    int d = deg[wave];
    const float ninf = -__builtin_inff();
    float m0 = ninf, m1 = ninf, m2 = ninf, m3 = ninf;
    int src = (d > 0) ? csr_src[beg] : 0;
    for (int e = 0; e < d; ++e) {
        int nsrc = (e + 1 < d) ? csr_src[beg + e + 1] : src;
        __builtin_prefetch(X + (size_t)nsrc * D_IN + (lane << 2), 0, 0);  // global_prefetch_b8
        const float4 xr = *(const float4*)(X + (size_t)src * D_IN + (lane << 2));
        m0 = fmaxf(m0, xr.x);
        m1 = fmaxf(m1, xr.y);
        m2 = fmaxf(m2, xr.z);
        m3 = fmaxf(m3, xr.w);
        src = nsrc;
    }
    if (d == 0) { m0 = 0.f; m1 = 0.f; m2 = 0.f; m3 = 0.f; }  // empty segment -> 0 (PyG fill)
    float4 o; o.x = m0; o.y = m1; o.z = m2; o.w = m3;
    *(float4*)(agg + (size_t)wave * D_IN + (lane << 2)) = o;
}

// ---------------- Pass 5: dual GEMV via V_WMMA_F32_16X16X4_F32 ----------------
// One wave32 per tile of 16 nodes. A (16x4 f32, documented layout):
//   lanes 0-15 hold rows M=0..15; VGPR0/1 = K{0,1}; lanes 16-31 = K{2,3}.
// B (4x16 f32): column N = lane&15; only N==0 carries the weight vector;
//   lane half selects K{0,1} vs K{2,3} (rows split across lane halves like C).
// D[:,0] accumulates out = agg@W_l^T + X@W_r^T over K=128 in chunks of 4.
__global__ void sage_proj_wmma(const float* __restrict__ agg, const float* __restrict__ X,
                               const float* __restrict__ Wl, const float* __restrict__ Wr,
                               const float* __restrict__ bl, float* __restrict__ out,
                               int n_nodes) {
    int wave = (blockIdx.x * blockDim.x + threadIdx.x) >> 5;
    int lane = threadIdx.x & 31;
    int tile = wave * 16;
    if (tile >= n_nodes) return;  // wave-uniform: EXEC stays all-ones for WMMA

    int row = tile + (lane & 15);
    int khalf = (lane >> 4) << 1;      // 0 for lanes 0-15, 2 for lanes 16-31
    bool col0 = (lane & 15) == 0;

    v8f acc = {};
    for (int k0 = 0; k0 < D_IN; k0 += 4) {
        const float* pa = agg + (size_t)row * D_IN + k0 + khalf;
        const float* px = X + (size_t)row * D_IN + k0 + khalf;
        v2f a;  a.x = pa[0]; a.y = pa[1];
        v2f ax; ax.x = px[0]; ax.y = px[1];
        float wl0 = Wl[k0 + khalf], wl1 = Wl[k0 + khalf + 1];
        float wr0 = Wr[k0 + khalf], wr1 = Wr[k0 + khalf + 1];
        v2f bwl; bwl.x = col0 ? wl0 : 0.f; bwl.y = col0 ? wl1 : 0.f;
        v2f bwr; bwr.x = col0 ? wr0 : 0.f; bwr.y = col0 ? wr1 : 0.f;
        acc = __builtin_amdgcn_wmma_f32_16x16x4_f32(false, a,  false, bwl, (short)0, acc, false, false);
        acc = __builtin_amdgcn_wmma_f32_16x16x4_f32(false, ax, false, bwr, (short)0, acc, false, false);
    }
    float b = bl[0];
    // C layout: column N=0 lives in lane 0 (M=0..7 in v0..v7) and lane 16 (M=8..15).
    if (lane == 0) {
        #pragma unroll
        for (int m = 0; m < 8; ++m) out[tile + m] = acc[m] + b;
    } else if (lane == 16) {
        #pragma unroll
        for (int m = 0; m < 8; ++m) out[tile + 8 + m] = acc[m] + b;
    }
}

extern "C" void kernel_launch(void* const* d_in, const int* in_sizes, int n_in,
                              void* d_out, int out_size, void* d_ws, size_t ws_size,
                              hipStream_t stream) {
    const float* X      = (const float*)d_in[0];
    const float* W_l    = (const float*)d_in[1];
    const float* b_l    = (const float*)d_in[2];
    const float* W_r    = (const float*)d_in[3];
    const long long* ei = (const long long*)d_in[4];
    float* out = (float*)d_out;

    const int n_nodes = in_sizes[0] / D_IN;   // 50000
    const int n_edges = in_sizes[4] / 2;      // 800000

    const int n_al = (n_nodes + 63) & ~63;
    const int e_al = (n_edges + 63) & ~63;
    int* deg    = (int*)d_ws;
    int* offs   = deg + n_al;
    int* cursor = offs + n_al;
    int* csr    = cursor + n_al;
    float* agg  = (float*)(csr + e_al);       // 25.6 MB, 16B-aligned

    sage_zero_deg<<<(n_nodes + 255) / 256, 256, 0, stream>>>(deg, n_nodes);
    sage_hist<<<(n_edges + 255) / 256, 256, 0, stream>>>(ei, deg, n_edges);
    sage_scan<<<1, 1024, 0, stream>>>(deg, offs, cursor, n_nodes);
    sage_fill<<<(n_edges + 255) / 256, 256, 0, stream>>>(ei, cursor, csr, n_edges);

    // one wave per node, 8 waves per block
    sage_maxagg<<<(n_nodes + 7) / 8, 256, 0, stream>>>(X, offs, deg, csr, agg, n_nodes);

    // one wave per 16-node tile
    const int tiles = (n_nodes + 15) / 16;
    sage_proj_wmma<<<(tiles + 7) / 8, 256, 0, stream>>>(agg, X, W_l, W_r, b_l, out, n_nodes);
}